// EIF_27513560498765
// MI455X (gfx1250) — compile-verified
//
#include <hip/hip_runtime.h>
#include <cstdint>

// ---------------------------------------------------------------------------
// EIF neuron scan: x[B,T,N] -> spikes[B,T,N], sequential over T per (b,n).
// Memory-bound (512 MiB traffic, ~22us floor @ 23.3 TB/s). Latency hiding via
// a 32-deep per-wave async global->LDS FIFO (CDNA5 ASYNCcnt path), no barriers.
// In-flight reads: 512 blocks x 2 waves x 32 x 512B ~= 16 MB ~= BDP @ 700ns.
// ---------------------------------------------------------------------------

#define BLOCK_T 64            // 2 wave32s per block

typedef __attribute__((ext_vector_type(4))) float v4f;  // native clang vector

constexpr int kB   = 32;
constexpr int kT   = 512;
constexpr int kN   = 4096;
constexpr int kVec = 4;                        // v4f per lane
constexpr int kTileN = BLOCK_T * kVec;         // 256 n-values per block
constexpr int kNBlk  = kN / kTileN;            // 16 blocks along N
constexpr int kDepth = 32;                     // async pipeline stages
static_assert(kDepth == 32, "WAIT_ASYNCCNT immediate below assumes depth 32");

#define EIF_V_TH   1.0f
#define EIF_DT     0.2f
// exp((m - V_TH)/DT) == exp2((m - V_TH) * log2(e)/DT); log2(e)/0.2 = 7.213475...
#define EIF_KEXP   7.2134752044448169f

// Issue one lane-private 16B async copy global -> LDS (tracked by ASYNCcnt).
__device__ __forceinline__ void async_copy_b128(uint32_t lds_addr, const float* gsrc) {
    asm volatile("global_load_async_to_lds_b128 %0, %1, off"
                 :
                 : "v"(lds_addr), "v"((uint64_t)(uintptr_t)gsrc)
                 : "memory");
}

// Wait until at most `n` async ops outstanding (in-order completion).
#define WAIT_ASYNCCNT(n) asm volatile("s_wait_asynccnt " #n ::: "memory")

// One EIF timestep for one scalar lane element. TAU == 1.
__device__ __forceinline__ float eif_step(float& m, float xv) {
    float e  = __builtin_amdgcn_exp2f((m - EIF_V_TH) * EIF_KEXP);  // v_exp_f32
    float mn = m + (xv - m + EIF_DT * e);                          // /TAU with TAU=1
    bool  sp = (mn >= EIF_V_TH);
    m = sp ? 0.0f : mn;                                            // V_RESET = 0
    return sp ? 1.0f : 0.0f;
}

__global__ __launch_bounds__(BLOCK_T)
void eif_scan_kernel(const float* __restrict__ x, float* __restrict__ out) {
    // 32 stages x 64 lanes x 16B = 32 KB LDS FIFO (per-lane private slots).
    __shared__ v4f stagebuf[kDepth * BLOCK_T];

    const int tid = threadIdx.x;
    const int b   = blockIdx.x / kNBlk;
    const int nb  = blockIdx.x % kNBlk;

    const long long col  = (long long)nb * kTileN + (long long)tid * kVec;
    const long long base = (long long)b * kT * kN + col;
    const float* gx = x   + base;
    float*       gy = out + base;

    // LDS byte address of this lane's stage-0 slot (low 32 bits of generic ptr).
    const uint32_t slot0        = (uint32_t)(uintptr_t)(const void*)&stagebuf[tid];
    const uint32_t stage_stride = BLOCK_T * (uint32_t)sizeof(v4f);

    // ---- prime the pipeline: rows 0..kDepth-1 ----
    const float* gload = gx;
    #pragma unroll
    for (int s = 0; s < kDepth; ++s) {
        async_copy_b128(slot0 + (uint32_t)s * stage_stride, gload);
        gload += kN;               // gload now -> row kDepth
    }

    float mem[kVec] = {0.0f, 0.0f, 0.0f, 0.0f};  // V_RESET
    float* gstore  = gy;
    uint32_t stage = 0;

    for (int t = 0; t < kT; ++t) {
        // Oldest of <=32 outstanding loads (row t) has landed in LDS.
        WAIT_ASYNCCNT(31);

        v4f xv = stagebuf[stage * BLOCK_T + (uint32_t)tid];  // ds_load_b128

        v4f sp;
        sp.x = eif_step(mem[0], xv.x);
        sp.y = eif_step(mem[1], xv.y);
        sp.z = eif_step(mem[2], xv.z);
        sp.w = eif_step(mem[3], xv.w);

        // Streaming output: nontemporal 16B store.
        __builtin_nontemporal_store(sp, (v4f*)gstore);
        gstore += kN;

        // Refill the freed stage with row t+kDepth (clamped in drain phase so
        // the constant-immediate wait above stays valid; clamped rows land in
        // stages that are never read again).
        async_copy_b128(slot0 + stage * stage_stride, gload);
        if (t + kDepth + 1 < kT) gload += kN;

        stage = (stage + 1) & (kDepth - 1);
    }
}

extern "C" void kernel_launch(void* const* d_in, const int* in_sizes, int n_in,
                              void* d_out, int out_size, void* d_ws, size_t ws_size,
                              hipStream_t stream) {
    (void)in_sizes; (void)n_in; (void)out_size; (void)d_ws; (void)ws_size;
    const float* x = (const float*)d_in[0];
    float* out = (float*)d_out;

    dim3 grid(kB * kNBlk);   // 512 blocks
    dim3 block(BLOCK_T);     // 64 threads (2 wave32s)
    eif_scan_kernel<<<grid, block, 0, stream>>>(x, out);
}